// _AnchorLayer_35871566856276
// MI455X (gfx1250) — compile-verified
//
#include <hip/hip_runtime.h>
#include <hip/hip_bf16.h>
#include <stdint.h>

// Problem constants from the reference (H=W=80, A=4 sizes {2,4,8,16}, K=64).
#define HH      80
#define WW      80
#define AA      4
#define KK      64
#define NANCH   (AA * HH * WW)   // 25600 unique anchors (replicated B times in output)

__global__ void _init_gtmax_kernel(float* __restrict__ gtmax, int n) {
    int i = blockIdx.x * blockDim.x + threadIdx.x;
    if (i < n) gtmax[i] = -__builtin_inff();
}

// Hardware v_rcp_f32 + one Newton step: ~0.5 ulp, 4 VALU ops instead of the
// ~10-op IEEE divide expansion. inf/0 denominators differ from the reference
// (inter/inf and inter/0 -> NaN via 0*inf in the Newton step; NaN propagates).
__device__ __forceinline__ float fast_div(float num, float den) {
    float r = __builtin_amdgcn_rcpf(den);
    r = r * __builtin_fmaf(-den, r, 2.0f);
    return num * r;
}

// BR > 0: compile-time replica count (fully unrolled replica stores, constant
// strides). BR == 0: generic runtime-B fallback.
template <int BR>
__global__ __launch_bounds__(256)
void _anchor_iou_kernel(const float* __restrict__ gt_boxes,  // (B, K, 5)
                        float* __restrict__ out,
                        int Brt, int Nall,
                        unsigned long long maxOff,
                        unsigned long long argOff,
                        unsigned long long gtOff)
{
    const int nrep = (BR > 0) ? BR : Brt;   // constant when BR > 0

    const int tid = threadIdx.x;
    const int b   = blockIdx.y;
    const int k4  = (tid & 15) * 4;   // this lane's 4 GT columns
    const int r   = tid >> 4;         // 16-lane row-group id (0..15)

    __shared__ __align__(16) float raw[KK * 5];           // staged gt_boxes[b]
    __shared__ float gx1[KK], gy1[KK], gx2[KK], gy2[KK], gar[KK], gzf[KK];
    __shared__ float kmax[KK];

    // --- CDNA5 async copy: gt_boxes[b] (320 floats) -> LDS, 80 lanes x b128 ---
    if (tid < 80) {
        unsigned int lds_off = (unsigned int)(uintptr_t)(&raw[tid * 4]);
        unsigned long long ga =
            (unsigned long long)(uintptr_t)(gt_boxes + (size_t)b * KK * 5 + tid * 4);
        asm volatile("global_load_async_to_lds_b128 %0, %1, off"
                     :: "v"(lds_off), "v"(ga) : "memory");
    }
    if (tid < KK) kmax[tid] = -__builtin_inff();
    asm volatile("s_wait_asynccnt 0" ::: "memory");
    __syncthreads();

    // Derived per-GT values (replicates reference formulas exactly)
    if (tid < KK) {
        float x1 = raw[tid * 5 + 0], y1 = raw[tid * 5 + 1];
        float x2 = raw[tid * 5 + 2], y2 = raw[tid * 5 + 3];
        float gx = x2 - x1 + 1.0f, gy = y2 - y1 + 1.0f;
        gx1[tid] = x1; gy1[tid] = y1; gx2[tid] = x2; gy2[tid] = y2;
        gar[tid] = gx * gy;
        gzf[tid] = (gx == 1.0f && gy == 1.0f) ? 1.0f : 0.0f;
    }
    __syncthreads();

    // Hoist this lane's 4 GT boxes into registers; inner loop is memory-free.
    float Gx1[4], Gy1[4], Gx2[4], Gy2[4], Gar[4], Gz[4];
#pragma unroll
    for (int j = 0; j < 4; ++j) {
        int k = k4 + j;
        Gx1[j] = gx1[k]; Gy1[j] = gy1[k]; Gx2[j] = gx2[k]; Gy2[j] = gy2[k];
        Gar[j] = gar[k]; Gz[j]  = gzf[k];
    }

    // Block covers 128 unique anchor rows; 128 | 6400 so the whole block sits
    // in one anchor-size slab -> size is a per-block scalar (SALU).
    const int   nBase   = blockIdx.x * 128;          // unique row base, < NANCH
    const int   aIdx    = nBase / (HH * WW);
    const int   remBase = nBase % (HH * WW);
    const float s       = (float)(2 << aIdx);        // {2,4,8,16}
    const float half    = 0.5f * s;
    const float bnd     = (float)(HH - 1);

    const size_t repStride = (size_t)NANCH * KK;     // replica stride in overlaps
    const size_t ovBase    = (size_t)b * (size_t)Nall * KK;
    float*  pOv  = out + ovBase + (size_t)(nBase + r) * KK + k4;
    float*  pMax = out + maxOff + (size_t)b * Nall + nBase + r;
    float*  pArg = out + argOff + (size_t)b * Nall + nBase + r;

    float gacc[4] = {-__builtin_inff(), -__builtin_inff(),
                     -__builtin_inff(), -__builtin_inff()};

#pragma unroll
    for (int i = 0; i < 8; ++i) {
        const int rem = remBase + i * 16 + r;
        const int yy  = rem / WW;                    // magic-mul
        const int xx  = rem - yy * WW;

        float x0 = (float)xx - half, y0 = (float)yy - half;
        float x2 = x0 + s,           y2 = y0 + s;
        float x0c = fminf(fmaxf(x0, 0.0f), bnd);
        float y0c = fminf(fmaxf(y0, 0.0f), bnd);
        float x2c = fminf(fmaxf(x2, 0.0f), bnd);
        float y2c = fminf(fmaxf(y2, 0.0f), bnd);
        // Reference stores [x0c, y0c, x2c-x0c, y2c-y0c] then treats cols 2,3 as
        // coordinates again — replicate exactly.
        float ax1 = x0c, ay1 = y0c;
        float ax2 = x2c - x0c, ay2 = y2c - y0c;
        float anx = ax2 - ax1 + 1.0f, any_ = ay2 - ay1 + 1.0f;
        float aar = anx * any_;
        bool  az  = (anx == 1.0f) && (any_ == 1.0f);

        float4 ovv;
        float best = 0.0f; int bestk = 0;
#pragma unroll
        for (int j = 0; j < 4; ++j) {
            float iw = fminf(ax2, Gx2[j]) - fmaxf(ax1, Gx1[j]) + 1.0f;
            iw = fmaxf(iw, 0.0f);
            float ih = fminf(ay2, Gy2[j]) - fmaxf(ay1, Gy1[j]) + 1.0f;
            ih = fmaxf(ih, 0.0f);
            float inter = iw * ih;
            float ua    = aar + Gar[j] - inter;
            float ov    = fast_div(inter, ua);
            ov = (Gz[j] != 0.0f) ? 0.0f : ov;
            ov = az ? -1.0f : ov;
            (&ovv.x)[j] = ov;
            if (j == 0 || ov > best) { best = ov; bestk = k4 + j; }  // first occurrence
            gacc[j] = fmaxf(gacc[j], ov);
        }

        // Anchors are batch-replicated: emit the same 16B to all replicas.
#pragma unroll
        for (int rep = 0; rep < nrep; ++rep)
            *(float4*)(pOv + (size_t)rep * repStride) = ovv;

        // Row max across the 16 lanes owning this row (wave32 shuffles),
        // then first-occurrence argmax via ballot (lanes are k-ordered).
        float bv = best;
#pragma unroll
        for (int m = 1; m < 16; m <<= 1) bv = fmaxf(bv, __shfl_xor(bv, m, 16));
        unsigned long long bm = __ballot(best == bv);
        unsigned grp = (unsigned)(bm >> (tid & 16)) & 0xFFFFu;
        int firstLane = __ffs(grp) - 1;
        int bk = __shfl(bestk, firstLane, 16);

        if ((tid & 15) == 0) {
            float bkf = (float)bk;
#pragma unroll
            for (int rep = 0; rep < nrep; ++rep) {
                pMax[i * 16 + rep * NANCH] = bv;
                pArg[i * 16 + rep * NANCH] = bkf;
            }
        }
        pOv += 16 * KK;
    }

    // gt_max: thread -> LDS float atomic max -> one global atomic per k per block.
#pragma unroll
    for (int j = 0; j < 4; ++j) atomicMax(&kmax[k4 + j], gacc[j]);
    __syncthreads();
    if (tid < KK) atomicMax(out + gtOff + (size_t)b * KK + tid, kmax[tid]);
}

extern "C" void kernel_launch(void* const* d_in, const int* in_sizes, int n_in,
                              void* d_out, int out_size, void* d_ws, size_t ws_size,
                              hipStream_t stream) {
    (void)n_in; (void)out_size; (void)d_ws; (void)ws_size;
    const float* gt = (const float*)d_in[1];          // gt_boxes (B,K,5)
    float* out = (float*)d_out;

    const int B    = in_sizes[1] / (KK * 5);          // 4
    const int Nall = B * NANCH;                       // 102400 (flattened incl. batch replicas)

    const unsigned long long ovSize = (unsigned long long)B * Nall * KK;
    const unsigned long long maxOff = ovSize;
    const unsigned long long argOff = maxOff + (unsigned long long)B * Nall;
    const unsigned long long gtOff  = argOff + (unsigned long long)B * Nall;

    _init_gtmax_kernel<<<1, 256, 0, stream>>>(out + gtOff, B * KK);

    dim3 grid(NANCH / 128, B);                        // 200 x B blocks, 256 thr (8 waves)
    if (B == 4) {
        _anchor_iou_kernel<4><<<grid, 256, 0, stream>>>(gt, out, B, Nall,
                                                        maxOff, argOff, gtOff);
    } else {
        _anchor_iou_kernel<0><<<grid, 256, 0, stream>>>(gt, out, B, Nall,
                                                        maxOff, argOff, gtOff);
    }
}